// Unified_Conv2d_Block_47639777247518
// MI455X (gfx1250) — compile-verified
//
#include <hip/hip_runtime.h>
#include <hip/hip_bf16.h>
#include <math.h>

typedef __attribute__((ext_vector_type(2))) float v2f;
typedef __attribute__((ext_vector_type(8))) float v8f;

#define BATCH 16
#define IC 32
#define OC 96
#define SW 64            // spatial width == height
#define KW 65            // stored kernel extent
#define NTAP 9
#define XS_CHP 17        // channel pairs per (row,col) cell, padded 16->17 (34 dwords)

// ---------------------------------------------------------------------------
// 1) Gather the 27,648 live (masked) weights directly into A-fragment order:
//    WcF[g][tap][mhalf][lane][cq*2+e], value = W[o, c, tap] with
//    o = g*32 + mhalf*16 + (lane&15),  c = cq*4 + (lane>>4)*2 + e.
//    One tap's fragments for a wave are 64B/lane contiguous -> b128 loads.
// ---------------------------------------------------------------------------
__global__ __launch_bounds__(256) void prep_weights_kernel(
    const float* __restrict__ weight, const float* __restrict__ mask,
    float* __restrict__ WcF) {
  int idx = blockIdx.x * 256 + threadIdx.x;            // 3*9*2*32*16 = 27648
  if (idx >= 3 * NTAP * 2 * 32 * 16) return;
  int e     = idx & 1;
  int cq    = (idx >> 1) & 7;
  int lane  = (idx >> 4) & 31;
  int mhalf = (idx >> 9) & 1;
  int t2    = idx >> 10;
  int tap   = t2 % NTAP;
  int g     = t2 / NTAP;
  int d     = g + 1;
  int ki    = 32 + (tap / 3 - 1) * d;
  int kj    = 32 + (tap % 3 - 1) * d;
  int oc    = g * 32 + mhalf * 16 + (lane & 15);
  int c     = cq * 4 + (lane >> 4) * 2 + e;
  float w = weight[((oc * IC + c) * KW + ki) * KW + kj];
  float m = mask[oc * (KW * KW) + ki * KW + kj];
  WcF[idx] = w * m;
}

// ---------------------------------------------------------------------------
// 2) Circular dilated 3x3 conv as fp32 WMMA GEMM.
//    Per tap: 4x global_load_b128 (A-frags) + 8x ds_load_b64 (B-frags)
//    + 8x v_wmma_f32_16x16x4_f32.  72 WMMAs per wave tile (16 out x 16 px).
// ---------------------------------------------------------------------------
__global__ __launch_bounds__(256) void conv_wmma_kernel(
    const float* __restrict__ x, const float* __restrict__ WcF,
    float* __restrict__ y) {
  // [rr 0..6][col 0..63][channel-pair 0..15, padded to 17]
  __shared__ v2f xs2[7 * SW * XS_CHP];   // 7*64*17*8B = 60,928 B

  const int tid = threadIdx.x;
  const int row = blockIdx.x;   // 0..63
  const int g   = blockIdx.y;   // 0..2  (dilation = g+1)
  const int n   = blockIdx.z;   // 0..15
  const int d   = g + 1;

  // Stage 7 circularly-wrapped rows x 32 channels, channel-pair layout.
  {
    const int col = tid & 63;
    const int q   = tid >> 6;                 // 0..3 -> channel pairs q*4..q*4+3
    const float* xp = x + ((n * IC) << 12) + col;
    for (int rr = 0; rr < 7; ++rr) {
      const int srow = (row + rr - 3 + SW) & 63;
      const float* xr = xp + (srow << 6);
      v2f* sp = xs2 + (rr * SW + col) * XS_CHP;
#pragma unroll
      for (int p8 = 0; p8 < 4; ++p8) {
        const int cp = q * 4 + p8;            // channel pair index 0..15
        v2f v;
        v.x = xr[(cp * 2) << 12];
        v.y = xr[(cp * 2 + 1) << 12];
        sp[cp] = v;
      }
    }
  }
  __syncthreads();

  const int wave  = tid >> 5;
  const int lane  = tid & 31;
  const int l16   = lane & 15;
  const int hi    = lane >> 4;          // selects K pair within quad
  const int mhalf = wave & 1;           // which 16 of the 32 out channels
  const int col0  = (wave >> 1) << 4;   // 4 col tiles x 16 = 64 cols

  v8f acc = {};

#pragma unroll
  for (int tap = 0; tap < NTAP; ++tap) {
    const int r_lds = 3 + (tap / 3 - 1) * d;
    const int coff  = (tap % 3 - 1) * d;
    const int cc    = (col0 + l16 + coff + SW) & 63;

    // A-fragments for all 8 k-steps of this tap: 16 contiguous floats/lane.
    const float4* wf4 = (const float4*)(WcF +
        (((g * NTAP + tap) * 2 + mhalf) * 32 + lane) * 16);
    float wv[16];
    *(float4*)(wv + 0)  = wf4[0];
    *(float4*)(wv + 4)  = wf4[1];
    *(float4*)(wv + 8)  = wf4[2];
    *(float4*)(wv + 12) = wf4[3];

    const v2f* xb = xs2 + (r_lds * SW + cc) * XS_CHP + hi;
#pragma unroll
    for (int cq = 0; cq < 8; ++cq) {
      v2f a;
      a.x = wv[cq * 2];
      a.y = wv[cq * 2 + 1];
      v2f b = xb[cq * 2];               // channels cq*4 + 2*hi, +1
      acc = __builtin_amdgcn_wmma_f32_16x16x4_f32(
          false, a, false, b, (short)0, acc, false, false);
    }
  }

  // D layout: VGPR r -> M = r (lanes 0-15) or r+8 (lanes 16-31), N = lane&15.
  const int col    = col0 + l16;
  const int ocbase = g * 32 + mhalf * 16 + hi * 8;
  float* out = y + ((n * OC + ocbase) << 12) + (row << 6) + col;
#pragma unroll
  for (int r = 0; r < 8; ++r) out[r << 12] = acc[r];
}

// ---------------------------------------------------------------------------
// 3) Per-channel batch statistics (mean, rsqrt(var+eps)), float4 reads,
//    deterministic LDS tree reduction.
// ---------------------------------------------------------------------------
__global__ __launch_bounds__(256) void stats_kernel(
    const float* __restrict__ y, float* __restrict__ mean,
    float* __restrict__ rsig) {
  __shared__ float ls[256];
  __shared__ float ls2[256];
  const int ch  = blockIdx.x;   // 0..95
  const int tid = threadIdx.x;
  float s = 0.f, s2 = 0.f;
  for (int i = tid; i < (BATCH * SW * SW) / 4; i += 256) {
    int nn = i >> 10;                       // 1024 float4 per image-channel
    int p4 = (i & 1023) << 2;
    float4 v = *(const float4*)(y + ((nn * OC + ch) << 12) + p4);
    s  += v.x + v.y + v.z + v.w;
    s2 += v.x * v.x + v.y * v.y + v.z * v.z + v.w * v.w;
  }
  ls[tid] = s; ls2[tid] = s2;
  __syncthreads();
  for (int off = 128; off > 0; off >>= 1) {
    if (tid < off) { ls[tid] += ls[tid + off]; ls2[tid] += ls2[tid + off]; }
    __syncthreads();
  }
  if (tid == 0) {
    const float inv = 1.0f / (float)(BATCH * SW * SW);
    float m  = ls[0] * inv;
    float v  = ls2[0] * inv - m * m;
    mean[ch] = m;
    rsig[ch] = rsqrtf(v + 1e-5f);
  }
}

// ---------------------------------------------------------------------------
// 4) Fused BatchNorm(affine) + tanh, in place, float4 vectorized.
// ---------------------------------------------------------------------------
__global__ __launch_bounds__(256) void bn_tanh_kernel(
    float* __restrict__ y, const float* __restrict__ mean,
    const float* __restrict__ rsig, const float* __restrict__ gamma,
    const float* __restrict__ beta) {
  int idx  = blockIdx.x * 256 + threadIdx.x;  // one float4 each; exact grid
  int base = idx << 2;
  int ch   = (base >> 12) % OC;
  float scale = rsig[ch] * gamma[ch];
  float shift = beta[ch] - mean[ch] * scale;
  float4 v = *(const float4*)(y + base);
  v.x = tanhf(v.x * scale + shift);
  v.y = tanhf(v.y * scale + shift);
  v.z = tanhf(v.z * scale + shift);
  v.w = tanhf(v.w * scale + shift);
  *(float4*)(y + base) = v;
}

// ---------------------------------------------------------------------------
extern "C" void kernel_launch(void* const* d_in, const int* in_sizes, int n_in,
                              void* d_out, int out_size, void* d_ws, size_t ws_size,
                              hipStream_t stream) {
  const float* x      = (const float*)d_in[0];
  const float* weight = (const float*)d_in[1];
  const float* mask   = (const float*)d_in[2];
  const float* gamma  = (const float*)d_in[3];
  const float* beta   = (const float*)d_in[4];
  float* y  = (float*)d_out;            // (16, 96, 64, 64)
  float* ws = (float*)d_ws;

  float* WcF  = ws;                      // 27648 floats, fragment-ordered
  float* mean = ws + 27648;              // 96 floats
  float* rsig = ws + 27648 + 96;         // 96 floats

  prep_weights_kernel<<<(3 * NTAP * 2 * 32 * 16 + 255) / 256, 256, 0, stream>>>(
      weight, mask, WcF);
  conv_wmma_kernel<<<dim3(SW, 3, BATCH), 256, 0, stream>>>(x, WcF, y);
  stats_kernel<<<OC, 256, 0, stream>>>(y, mean, rsig);
  const int total4 = BATCH * OC * SW * SW / 4;  // 1,572,864
  bn_tanh_kernel<<<total4 / 256, 256, 0, stream>>>(y, mean, rsig, gamma, beta);
}